// mlp2_64037962383751
// MI455X (gfx1250) — compile-verified
//
#include <hip/hip_runtime.h>

#define B_    8
#define CH    64
#define NPTS  4096
#define KNN   20
#define EPSV  1e-5f
#define CHUNK 1024
#define ROWS  16

typedef __attribute__((ext_vector_type(2))) float v2f;
typedef __attribute__((ext_vector_type(8))) float v8f;

__device__ __forceinline__ v8f wmma_f32_16x16x4(v2f a, v2f b, v8f c) {
  // D = A(16x4,f32) * B(4x16,f32) + C(16x16,f32)  -> v_wmma_f32_16x16x4_f32
  return __builtin_amdgcn_wmma_f32_16x16x4_f32(false, a, false, b, (short)0, c,
                                               false, false);
}

// ---------------------------------------------------------------- xx = sum x^2
__global__ __launch_bounds__(256) void sq_kernel(const float* __restrict__ x,
                                                 float* __restrict__ xx) {
  const int b = blockIdx.y;
  const int n = blockIdx.x * 256 + threadIdx.x;
  const float* xb = x + (size_t)b * CH * NPTS;
  float s = 0.f;
#pragma unroll
  for (int c = 0; c < CH; ++c) {
    float v = xb[(size_t)c * NPTS + n];
    s = fmaf(v, v, s);
  }
  xx[(size_t)b * NPTS + n] = s;
}

// ----------------------------------------- fused distance GEMM (WMMA) + top-20
__global__ __launch_bounds__(512) void knn_kernel(const float* __restrict__ x,
                                                  const float* __restrict__ xx,
                                                  int* __restrict__ idxout) {
  __shared__ float sbuf[ROWS * CHUNK];  // 64 KB neg-dist staging tile
  const int b = blockIdx.y, n0 = blockIdx.x * ROWS;
  const int t = threadIdx.x, lane = t & 31, wave = t >> 5;
  const int half = lane >> 4, l16 = lane & 15;
  const float* xb = x + (size_t)b * CH * NPTS;
  const float* xxb = xx + (size_t)b * NPTS;

  // Preload A operands (16 rows x 64 ch) in WMMA 16x4 f32 lane layout.
  float a0[16], a1[16];
#pragma unroll
  for (int kc = 0; kc < 16; ++kc) {
    int k0 = kc * 4 + half * 2;
    a0[kc] = xb[(size_t)k0 * NPTS + n0 + l16];
    a1[kc] = xb[(size_t)(k0 + 1) * NPTS + n0 + l16];
  }
  float xxr[8];
#pragma unroll
  for (int j = 0; j < 8; ++j) xxr[j] = xxb[n0 + j + 8 * half];

  // Per-lane top-20 (sorted desc), static indexing only -> stays in VGPRs.
  float tv[KNN]; int ti[KNN];
#pragma unroll
  for (int i = 0; i < KNN; ++i) { tv[i] = -3.4e38f; ti[i] = 0x7fffffff; }

  for (int chf = 0; chf < 4; ++chf) {
    const int cbase = chf * CHUNK;
    // --- GEMM phase: 64 col-tiles per chunk, 4 per wave, K=64 in 16 WMMAs ---
    for (int tt = 0; tt < 4; ++tt) {
      const int c0 = cbase + (tt * 16 + wave) * 16;
      v8f acc = {0.f, 0.f, 0.f, 0.f, 0.f, 0.f, 0.f, 0.f};
#pragma unroll
      for (int kc = 0; kc < 16; ++kc) {
        int k0 = kc * 4 + half * 2;
        v2f A = {a0[kc], a1[kc]};
        v2f Bv = {xb[(size_t)k0 * NPTS + c0 + l16],
                  xb[(size_t)(k0 + 1) * NPTS + c0 + l16]};
        acc = wmma_f32_16x16x4(A, Bv, acc);
      }
      const float xxc = xxb[c0 + l16];
      const int colLocal = c0 - cbase + l16;
#pragma unroll
      for (int j = 0; j < 8; ++j) {
        int rloc = j + 8 * half;  // C/D layout: M = j + 8*(lane>=16), N = lane%16
        sbuf[rloc * CHUNK + colLocal] = 2.f * acc[j] - xxr[j] - xxc;
      }
    }
    __syncthreads();
    // --- top-k phase: one wave per row, lane-strided, threshold + bubble ---
    {
      const float* rowp = &sbuf[wave * CHUNK];
#pragma unroll 1
      for (int i = 0; i < CHUNK / 32; ++i) {
        int cl = lane + i * 32;
        float v = rowp[cl];
        if (v > tv[KNN - 1]) {
          float cv = v; int ci = cbase + cl;
#pragma unroll
          for (int s = 0; s < KNN; ++s) {
            bool gt = (cv > tv[s]);  // strict: equal keeps earlier index first
            float nv = gt ? cv : tv[s]; int ni = gt ? ci : ti[s];
            float ov = gt ? tv[s] : cv; int oi = gt ? ti[s] : ci;
            tv[s] = nv; ti[s] = ni; cv = ov; ci = oi;
          }
        }
      }
    }
    __syncthreads();
  }

  // --- cross-lane merge: select 20 global winners via butterfly argmax ---
  const int outbase = ((b * NPTS) + n0 + wave) * KNN;
  for (int it = 0; it < KNN; ++it) {
    float bv = tv[0]; int bi = ti[0]; int bl = lane;
#pragma unroll
    for (int off = 16; off >= 1; off >>= 1) {
      float ov = __shfl_xor(bv, off, 32);
      int oi = __shfl_xor(bi, off, 32);
      int ol = __shfl_xor(bl, off, 32);
      bool take = (ov > bv) || (ov == bv && oi < bi);  // jax tie: smaller idx
      if (take) { bv = ov; bi = oi; bl = ol; }
    }
    if (lane == 0) idxout[outbase + it] = bi;
    if (lane == bl) {  // winner lane pops its head (static-index shift)
#pragma unroll
      for (int s = 0; s < KNN - 1; ++s) { tv[s] = tv[s + 1]; ti[s] = ti[s + 1]; }
      tv[KNN - 1] = -3.4e38f; ti[KNN - 1] = 0x7fffffff;
    }
  }
}

// ------------------- u = wA @ X, v = (wB - wA) @ X  (WMMA f32, transposed out)
__global__ __launch_bounds__(256) void uv_kernel(const float* __restrict__ x,
                                                 const float* __restrict__ w,
                                                 float* __restrict__ uT,
                                                 float* __restrict__ vT) {
  const int b = blockIdx.z, o0 = blockIdx.y * 16;
  const int lane = threadIdx.x & 31, wave = threadIdx.x >> 5;
  const int half = lane >> 4, l16 = lane & 15;
  const int c0 = blockIdx.x * 128 + wave * 16;
  const float* xb = x + (size_t)b * CH * NPTS;
  v8f au = {0.f, 0.f, 0.f, 0.f, 0.f, 0.f, 0.f, 0.f};
  v8f av = {0.f, 0.f, 0.f, 0.f, 0.f, 0.f, 0.f, 0.f};
#pragma unroll
  for (int kc = 0; kc < 16; ++kc) {
    int k0 = kc * 4 + half * 2;
    float wa0 = w[(o0 + l16) * 128 + k0];
    float wa1 = w[(o0 + l16) * 128 + k0 + 1];
    v2f A1 = {wa0, wa1};
    v2f A2 = {w[(o0 + l16) * 128 + 64 + k0] - wa0,
              w[(o0 + l16) * 128 + 64 + k0 + 1] - wa1};
    v2f Bv = {xb[(size_t)k0 * NPTS + c0 + l16],
              xb[(size_t)(k0 + 1) * NPTS + c0 + l16]};
    au = wmma_f32_16x16x4(A1, Bv, au);
    av = wmma_f32_16x16x4(A2, Bv, av);
  }
#pragma unroll
  for (int j = 0; j < 8; ++j) {
    int o = o0 + j + 8 * half;
    int n = c0 + l16;
    size_t oidx = ((size_t)b * NPTS + n) * 64 + o;
    uT[oidx] = au[j];
    vT[oidx] = av[j];
  }
}

// ------------- gather s = u[idx]+v, per-(b,o,n) max/min over k + BN moments
__global__ __launch_bounds__(256) void gather_kernel(
    const float* __restrict__ uT, const float* __restrict__ vT,
    const int* __restrict__ idx, float* __restrict__ Mmax,
    float* __restrict__ Mmin, float* __restrict__ stats) {
  __shared__ float bsum[64], bsq[64];
  const int b = blockIdx.y;
  const int t = threadIdx.x, lane = t & 31, wave = t >> 5;
  if (t < 64) { bsum[t] = 0.f; bsq[t] = 0.f; }
  __syncthreads();
  const size_t bN = (size_t)b * NPTS;
  const int wg = blockIdx.x * 8 + wave;  // 512 waves per batch
  float ls0 = 0.f, ls1 = 0.f, lq0 = 0.f, lq1 = 0.f;
  for (int i = 0; i < 8; ++i) {
    const int n = wg + 512 * i;
    float vv0 = vT[(bN + n) * 64 + lane];
    float vv1 = vT[(bN + n) * 64 + lane + 32];
    int mi = 0;
    if (lane < KNN) mi = idx[(bN + n) * KNN + lane];
    float mx0 = -3.4e38f, mx1 = -3.4e38f, mn0 = 3.4e38f, mn1 = 3.4e38f;
#pragma unroll
    for (int k = 0; k < KNN; ++k) {
      int j = __shfl(mi, k, 32);
      float s0 = uT[(bN + j) * 64 + lane] + vv0;       // 256B coalesced gather
      float s1 = uT[(bN + j) * 64 + lane + 32] + vv1;
      mx0 = fmaxf(mx0, s0); mn0 = fminf(mn0, s0); ls0 += s0; lq0 += s0 * s0;
      mx1 = fmaxf(mx1, s1); mn1 = fminf(mn1, s1); ls1 += s1; lq1 += s1 * s1;
    }
    Mmax[(bN + n) * 64 + lane] = mx0; Mmax[(bN + n) * 64 + lane + 32] = mx1;
    Mmin[(bN + n) * 64 + lane] = mn0; Mmin[(bN + n) * 64 + lane + 32] = mn1;
  }
  atomicAdd(&bsum[lane], ls0); atomicAdd(&bsum[lane + 32], ls1);
  atomicAdd(&bsq[lane], lq0);  atomicAdd(&bsq[lane + 32], lq1);
  __syncthreads();
  if (t < 64) {
    atomicAdd(&stats[t], bsum[t]);
    atomicAdd(&stats[64 + t], bsq[t]);
  }
}

// -------------------------------------------------- BN scale/shift from moments
__global__ void finalize_kernel(const float* __restrict__ stats,
                                const float* __restrict__ g,
                                const float* __restrict__ bb,
                                float* __restrict__ params) {
  int o = threadIdx.x;
  if (o < 64) {
    const float cnt = (float)B_ * (float)NPTS * (float)KNN;
    float mean = stats[o] / cnt;
    float var = stats[64 + o] / cnt - mean * mean;
    float sc = g[o] * rsqrtf(var + EPSV);
    params[o] = sc;
    params[64 + o] = bb[o] - mean * sc;
  }
}

// ------------- y[b][o][n] = (relu?) (sc * (sc>=0 ? Mmax : Mmin) + sh), transpose
__global__ __launch_bounds__(256) void transform_kernel(
    const float* __restrict__ Mmax, const float* __restrict__ Mmin,
    const float* __restrict__ params, float* __restrict__ out, int relu) {
  __shared__ float tile[64][65];
  const int b = blockIdx.y, n0 = blockIdx.x * 64;
  const int t = threadIdx.x, tx = t & 63, tz = t >> 6;
  const size_t bN = (size_t)b * NPTS;
  const float sc = params[tx], sh = params[64 + tx];
  const float* src = (sc >= 0.f) ? Mmax : Mmin;  // monotone BN+max_k shortcut
  for (int r = tz; r < 64; r += 4) {
    float v = fmaf(sc, src[(bN + n0 + r) * 64 + tx], sh);
    if (relu) v = fmaxf(v, 0.f);
    tile[r][tx] = v;
  }
  __syncthreads();
  for (int o = tz; o < 64; o += 4)
    out[((size_t)b * 64 + o) * NPTS + n0 + tx] = tile[tx][o];
}

extern "C" void kernel_launch(void* const* d_in, const int* in_sizes, int n_in,
                              void* d_out, int out_size, void* d_ws,
                              size_t ws_size, hipStream_t stream) {
  (void)in_sizes; (void)n_in; (void)out_size; (void)ws_size;
  const float* x  = (const float*)d_in[0];
  const float* w1 = (const float*)d_in[1];
  const float* g1 = (const float*)d_in[2];
  const float* b1 = (const float*)d_in[3];
  const float* w2 = (const float*)d_in[4];
  const float* g2 = (const float*)d_in[5];
  const float* b2 = (const float*)d_in[6];
  float* out = (float*)d_out;

  char* ws = (char*)d_ws;
  float* xx   = (float*)ws;                       // 32768 f  (128 KB)
  int*   idx  = (int*)(ws + 131072);              // 655360 i (2.5 MB)
  float* uT   = (float*)(ws + 131072 + 2621440);  // 2M f (8 MB)
  float* vT   = uT + 2097152;                     // 8 MB
  float* Mmax = vT + 2097152;                     // 8 MB
  float* Mmin = Mmax + 2097152;                   // 8 MB
  float* h    = Mmin + 2097152;                   // 8 MB
  float* stats  = h + 2097152;                    // 128 f
  float* params = stats + 128;                    // 128 f

  auto layer = [&](const float* xin, const float* w, const float* g,
                   const float* bb, float* yout, int relu) {
    sq_kernel<<<dim3(NPTS / 256, B_), 256, 0, stream>>>(xin, xx);
    knn_kernel<<<dim3(NPTS / ROWS, B_), 512, 0, stream>>>(xin, xx, idx);
    uv_kernel<<<dim3(32, 4, B_), 256, 0, stream>>>(xin, w, uT, vT);
    hipMemsetAsync(stats, 0, 128 * sizeof(float), stream);
    gather_kernel<<<dim3(64, B_), 256, 0, stream>>>(uT, vT, idx, Mmax, Mmin,
                                                    stats);
    finalize_kernel<<<1, 64, 0, stream>>>(stats, g, bb, params);
    transform_kernel<<<dim3(NPTS / 64, B_), 256, 0, stream>>>(Mmax, Mmin,
                                                              params, yout,
                                                              relu);
  };

  layer(x, w1, g1, b1, h, 1);     // EdgeConv block 1 (relu(max))
  layer(h, w2, g2, b2, out, 0);   // EdgeConv block 2 (max only)
}